// SurrogateGraphSAGE_85985245266463
// MI455X (gfx1250) — compile-verified
//
#include <hip/hip_runtime.h>
#include <math.h>

#define NN 50000
#define NE 800000
#define D  128
#define DOUT 40

typedef __attribute__((ext_vector_type(2))) float v2f;
typedef __attribute__((ext_vector_type(8))) float v8f;

// ---------------- edge scatter: agg += x[src] (per dst), deg += 1 ----------------
__global__ __launch_bounds__(256) void sage_scatter(const long long* __restrict__ src,
                                                    const long long* __restrict__ dst,
                                                    const float* __restrict__ x,
                                                    float* __restrict__ agg,
                                                    float* __restrict__ deg) {
  unsigned tid  = blockIdx.x * 256u + threadIdx.x;
  unsigned e    = tid >> 5;
  unsigned lane = tid & 31u;
  if (e >= NE) return;
  int s = (int)src[e];
  int d = (int)dst[e];
  const float4 v = *(const float4*)(x + (size_t)s * D + lane * 4);
  float* o = agg + (size_t)d * D + lane * 4;
  unsafeAtomicAdd(o + 0, v.x);
  unsafeAtomicAdd(o + 1, v.y);
  unsafeAtomicAdd(o + 2, v.z);
  unsafeAtomicAdd(o + 3, v.w);
  if (lane == 0) unsafeAtomicAdd(deg + d, 1.0f);
}

// ---------------- agg /= max(deg,1) ----------------
__global__ __launch_bounds__(256) void sage_norm(float* __restrict__ agg,
                                                 const float* __restrict__ deg) {
  unsigned tid  = blockIdx.x * 256u + threadIdx.x;   // NN*32 threads
  unsigned node = tid >> 5;
  unsigned c    = (tid & 31u) << 2;
  float dg = deg[node];
  float inv = 1.0f / (dg < 1.0f ? 1.0f : dg);
  float4* p = (float4*)(agg + (size_t)node * D + c);
  float4 v = *p;
  v.x *= inv; v.y *= inv; v.z *= inv; v.w *= inv;
  *p = v;
}

// ---------------- one SAGE layer: h = relu(aggm @ Wl^T + bl + xin @ Wr^T) ----------------
// block = 256 thr = 8 waves; block -> 16-node tile, wave w -> cols [16w,16w+16)
#define LDSTRIDE 132   // 128 + 4 dword pad: lanes 0..15 hit banks 4r+kb (all distinct)
__global__ __launch_bounds__(256) void sage_gemm(const float* __restrict__ xin,
                                                 const float* __restrict__ aggm,
                                                 const float* __restrict__ Wl,
                                                 const float* __restrict__ bl,
                                                 const float* __restrict__ Wr,
                                                 float* __restrict__ hout) {
  __shared__ float sA[16 * LDSTRIDE];
  __shared__ float sX[16 * LDSTRIDE];
  const int m0 = blockIdx.x * 16;
  const int t  = threadIdx.x;

  // cooperative stage of the 16x128 A tiles (8 floats per thread per matrix)
  for (int i = t * 4; i < 16 * D; i += 256 * 4) {
    int r = i >> 7, c = i & 127;
    *(float4*)&sA[r * LDSTRIDE + c] = *(const float4*)(aggm + (size_t)(m0 + r) * D + c);
    *(float4*)&sX[r * LDSTRIDE + c] = *(const float4*)(xin  + (size_t)(m0 + r) * D + c);
  }
  if (m0 + 16 < NN)  // speculative prefetch of next tile -> global_prefetch_b8
    __builtin_prefetch(aggm + (size_t)(m0 + 16) * D + (t & 31) * 16, 0, 1);
  __syncthreads();

  const int wave  = t >> 5;
  const int lane  = t & 31;
  const int n0    = wave * 16;
  const int col   = lane & 15;       // output col within tile (B/C/D lane)
  const int arow  = lane & 15;       // A row for this lane
  const int khalf = (lane >> 4) << 1;

  float bias = bl[n0 + col];
  v8f acc = {bias, bias, bias, bias, bias, bias, bias, bias};

  const float* wlrow = Wl + (size_t)(n0 + col) * D;  // B[k][col] = W[col][k]
  const float* wrrow = Wr + (size_t)(n0 + col) * D;

#pragma unroll
  for (int k0 = 0; k0 < D; k0 += 4) {
    int kb = k0 + khalf;
    v2f a = *(const v2f*)&sA[arow * LDSTRIDE + kb];
    v2f b = *(const v2f*)(wlrow + kb);
    acc = __builtin_amdgcn_wmma_f32_16x16x4_f32(false, a, false, b, (short)0, acc, false, false);
  }
#pragma unroll
  for (int k0 = 0; k0 < D; k0 += 4) {
    int kb = k0 + khalf;
    v2f a = *(const v2f*)&sX[arow * LDSTRIDE + kb];
    v2f b = *(const v2f*)(wrrow + kb);
    acc = __builtin_amdgcn_wmma_f32_16x16x4_f32(false, a, false, b, (short)0, acc, false, false);
  }

  const int rbase = m0 + ((lane >> 4) << 3);  // C layout: VGPR r -> row r (+8 for hi lanes)
#pragma unroll
  for (int r = 0; r < 8; ++r) {
    float v = acc[r];
    hout[(size_t)(rbase + r) * D + n0 + col] = v > 0.0f ? v : 0.0f;
  }
}

// ---------------- classifier + log_softmax ----------------
// block = 96 thr = 3 waves; 3 x 16 = 48 cols (40 valid), K = 256 (h1|h2 concat)
#define HSTRIDE 260
__global__ __launch_bounds__(96) void sage_head(const float* __restrict__ h1,
                                                const float* __restrict__ h2,
                                                const float* __restrict__ Wlin,
                                                const float* __restrict__ blin,
                                                float* __restrict__ out) {
  __shared__ float sH[16 * HSTRIDE];
  __shared__ float sL[16 * 48];
  const int m0 = blockIdx.x * 16;
  const int t  = threadIdx.x;

  for (int i = t; i < 16 * 256; i += 96) {
    int r = i >> 8, c = i & 255;
    float v = (c < 128) ? h1[(size_t)(m0 + r) * D + c]
                        : h2[(size_t)(m0 + r) * D + (c - 128)];
    sH[r * HSTRIDE + c] = v;
  }
  __syncthreads();

  const int wave  = t >> 5;
  const int lane  = t & 31;
  const int n0    = wave * 16;
  const int col   = n0 + (lane & 15);
  const int arow  = lane & 15;
  const int khalf = (lane >> 4) << 1;
  const bool valid = col < DOUT;
  const float mask = valid ? 1.0f : 0.0f;

  float bias = valid ? blin[col] : 0.0f;
  v8f acc = {bias, bias, bias, bias, bias, bias, bias, bias};
  const float* wrow = Wlin + (size_t)(valid ? col : 0) * 256;

#pragma unroll
  for (int k0 = 0; k0 < 256; k0 += 4) {
    int kb = k0 + khalf;
    v2f a = *(const v2f*)&sH[arow * HSTRIDE + kb];
    v2f b = *(const v2f*)(wrow + kb);
    b[0] *= mask; b[1] *= mask;   // zero invalid cols without EXEC divergence
    acc = __builtin_amdgcn_wmma_f32_16x16x4_f32(false, a, false, b, (short)0, acc, false, false);
  }

  const int rbase = (lane >> 4) << 3;
#pragma unroll
  for (int r = 0; r < 8; ++r)
    sL[(rbase + r) * 48 + n0 + (lane & 15)] = acc[r];
  __syncthreads();

  if (t < 16) {   // per-row log_softmax over 40 logits
    const float* L = &sL[t * 48];
    float m = -1e30f;
    for (int o = 0; o < DOUT; ++o) m = fmaxf(m, L[o]);
    float s = 0.0f;
    for (int o = 0; o < DOUT; ++o) s += __expf(L[o] - m);
    float ls = __logf(s);
    float* orow = out + (size_t)(m0 + t) * DOUT;
    for (int o = 0; o < DOUT; ++o) orow[o] = L[o] - m - ls;
  }
}

// ---------------- host launcher ----------------
extern "C" void kernel_launch(void* const* d_in, const int* in_sizes, int n_in,
                              void* d_out, int out_size, void* d_ws, size_t ws_size,
                              hipStream_t stream) {
  const float*     x    = (const float*)d_in[0];
  const long long* ei   = (const long long*)d_in[1];
  const float*     W1l  = (const float*)d_in[2];
  const float*     b1l  = (const float*)d_in[3];
  const float*     W1r  = (const float*)d_in[4];
  const float*     W2l  = (const float*)d_in[5];
  const float*     b2l  = (const float*)d_in[6];
  const float*     W2r  = (const float*)d_in[7];
  const float*     Wlin = (const float*)d_in[8];
  const float*     blin = (const float*)d_in[9];
  const long long* src  = ei;
  const long long* dst  = ei + NE;
  float* outp = (float*)d_out;

  // workspace carve: agg[NN*128] | deg[NN] | h1[NN*128] | h2[NN*128]  (~77 MB)
  float* agg = (float*)d_ws;
  float* deg = agg + (size_t)NN * D;
  float* h1  = deg + NN;
  float* h2  = h1 + (size_t)NN * D;

  const int scatter_blocks = (NE * 32) / 256;   // 100000 exact
  const int norm_blocks    = (NN * 32) / 256;   // 6250 exact
  const int tile_blocks    = NN / 16;           // 3125 exact

  // ---- layer 1 ----
  hipMemsetAsync(agg, 0, (size_t)NN * D * sizeof(float) + NN * sizeof(float), stream);
  sage_scatter<<<scatter_blocks, 256, 0, stream>>>(src, dst, x, agg, deg);
  sage_norm<<<norm_blocks, 256, 0, stream>>>(agg, deg);
  sage_gemm<<<tile_blocks, 256, 0, stream>>>(x, agg, W1l, b1l, W1r, h1);

  // ---- layer 2 ----
  hipMemsetAsync(agg, 0, (size_t)NN * D * sizeof(float) + NN * sizeof(float), stream);
  sage_scatter<<<scatter_blocks, 256, 0, stream>>>(src, dst, h1, agg, deg);
  sage_norm<<<norm_blocks, 256, 0, stream>>>(agg, deg);
  sage_gemm<<<tile_blocks, 256, 0, stream>>>(h1, agg, W2l, b2l, W2r, h2);

  // ---- head ----
  sage_head<<<tile_blocks, 96, 0, stream>>>(h1, h2, Wlin, blin, outp);
}